// Multiplication_11115375362059
// MI455X (gfx1250) — compile-verified
//
#include <hip/hip_runtime.h>
#include <hip/hip_bf16.h>

typedef __attribute__((ext_vector_type(2))) float v2f;
typedef __attribute__((ext_vector_type(4))) float v4f;
typedef __attribute__((ext_vector_type(8))) float v8f;

#define NROW 36   // K (padded): 9 (1x1 outer) + 25 (2x2 outer) + 2 pad
#define NCOL 48   // N (padded): 34 real output columns + 14 zero pad

// ---------------------------------------------------------------------------
// Init kernel: exact e3nn CG tables in double precision (Racah formula +
// real<->complex spherical-harmonic basis change), one thread per W entry.
// W stored column-major: Wcm[col*NROW + row].
// ---------------------------------------------------------------------------
__device__ __forceinline__ double dfact(int n){ double r=1.0; for(int i=2;i<=n;++i) r*=i; return r; }
__device__ __forceinline__ int imax(int a,int b){ return a>b?a:b; }
__device__ __forceinline__ int imin(int a,int b){ return a<b?a:b; }

__device__ double su2cg(int j1,int m1,int j2,int m2,int j3,int m3){
  if (m1+m2 != m3) return 0.0;
  int vmin = imax(imax(-j1+j2+m3, -j1+m1), 0);
  int vmax = imin(imin(j2+j3+m1, j3-j1+j2), j3+m3);
  if (vmax < vmin) return 0.0;
  double C = sqrt((2.0*j3+1.0)*dfact(j3+j1-j2)*dfact(j3-j1+j2)*dfact(j1+j2-j3)
                  *dfact(j3+m3)*dfact(j3-m3)
                  /(dfact(j1+j2+j3+1)*dfact(j1-m1)*dfact(j1+m1)*dfact(j2-m2)*dfact(j2+m2)));
  double S = 0.0;
  for (int v=vmin; v<=vmax; ++v){
    double sgn = ((v+j2+m2)&1) ? -1.0 : 1.0;
    S += sgn * dfact(j2+j3+m1-v)*dfact(j1-m1+v)
         / (dfact(v)*dfact(j3-j1+j2-v)*dfact(j3+m3-v)*dfact(v+j1-j2-m3));
  }
  return C*S;
}

__device__ void qent(int l, int r, int c, double& re, double& im){
  // row r (= l+m), col c of e3nn real->complex change of basis, degree l
  const double is2 = 0.7071067811865475244;
  int m = r - l;
  double vr=0.0, vi=0.0;
  if (m < 0){
    if (c == l - m) vr = is2;
    else if (c == l + m) vi = -is2;
  } else if (m == 0){
    if (c == l) vr = 1.0;
  } else {
    double s = (m & 1) ? -1.0 : 1.0;
    if (c == l + m) vr = s*is2;
    else if (c == l - m) vi = s*is2;
  }
  switch (l & 3){ // multiply by (-i)^l
    case 0: re =  vr; im =  vi; break;
    case 1: re =  vi; im = -vr; break;
    case 2: re = -vr; im = -vi; break;
    default: re = -vi; im =  vr; break;
  }
}

__device__ double w3j_real(int l, int la, int i, int j, int k){
  // real-basis wigner3j(l, la, la)[i][j][k], e3nn convention
  int n1 = 2*l+1, n2 = 2*la+1;
  double inv = 1.0 / sqrt(2.0*la+1.0);
  double acc = 0.0;
  for (int a=0; a<n1; ++a){
    double r1, i1; qent(l, a, i, r1, i1);
    if (r1==0.0 && i1==0.0) continue;
    for (int bq=0; bq<n2; ++bq){
      double r2, i2; qent(la, bq, j, r2, i2);
      if (r2==0.0 && i2==0.0) continue;
      int m3 = (a-l) + (bq-la);
      if (m3 < -la || m3 > la) continue;
      double r3, i3; qent(la, la+m3, k, r3, i3);
      if (r3==0.0 && i3==0.0) continue;
      i3 = -i3; // conj
      double cg = su2cg(l, a-l, la, bq-la, la, m3) * inv;
      double tr = r1*r2 - i1*i2;
      double ti = r1*i2 + i1*r2;
      acc += (tr*r3 - ti*i3) * cg;
    }
  }
  return acc;
}

__global__ void build_w(float* __restrict__ Wcm){
  int idx = blockIdx.x*blockDim.x + threadIdx.x;
  if (idx >= NROW*NCOL) return;
  int col = idx / NROW;
  int row = idx - col*NROW;
  double val = 0.0;
  if (col < 9 && row < 9){                       // 1x1 block: rows j*3+k, cols l/i
    int l = (col==0) ? 0 : (col<4 ? 1 : 2);
    int base = (l==0)?0:(l==1)?1:4;
    val = w3j_real(l, 1, col-base, row/3, row%3) * sqrt(2.0*l+1.0);
  } else if (col>=9 && col<34 && row>=9 && row<34){ // 2x2 block
    int c2 = col-9, r2 = row-9;
    int l = (c2==0)?0 : (c2<4)?1 : (c2<9)?2 : (c2<16)?3 : 4;
    int base = (l==0)?0:(l==1)?1:(l==2)?4:(l==3)?9:16;
    val = w3j_real(l, 2, c2-base, r2/5, r2%5) * sqrt(2.0*l+1.0);
  }
  Wcm[idx] = (float)val;
}

// ---------------------------------------------------------------------------
// Output column -> (segment offset within row, multiplicity stride) LUT.
// Row layout of out (9472 f32): [0e*1e:768][l0:256][l1:768][l2:1280]
//                               [l0:256][l1:768][l2:1280][l3:1792][l4:2304]
// ---------------------------------------------------------------------------
__constant__ int cOff[34] = {
  768,
  1024,1025,1026,
  1792,1793,1794,1795,1796,
  3072,
  3328,3329,3330,
  4096,4097,4098,4099,4100,
  5376,5377,5378,5379,5380,5381,5382,
  7168,7169,7170,7171,7172,7173,7174,7175,7176
};
__constant__ int cStr[34] = {
  1,3,3,3,5,5,5,5,5,
  1,3,3,3,5,5,5,5,5,7,7,7,7,7,7,7,9,9,9,9,9,9,9,9,9
};

// ---------------------------------------------------------------------------
// Main kernel: 1 block per batch row (256 threads = 256 multiplicities,
// 8 wave32). Each wave: stage u[36] per row to LDS, then
// OUT[32x34] = U[32x36] @ W[36x48] with V_WMMA_F32_16X16X4_F32.
// ---------------------------------------------------------------------------
__global__ __launch_bounds__(256)
void tp_wmma_kernel(const float* __restrict__ f1, const float* __restrict__ f2,
                    const float* __restrict__ Wcm, float* __restrict__ out){
  __shared__ float lds[256*NROW];   // 36 KB
  const int m     = threadIdx.x;
  const int b     = blockIdx.x;
  const int lane  = m & 31;
  const int wbase = m & ~31;
  const int bf1 = b*2304, bf2 = b*2816, bo = b*9472;

  float s = f1[bf1 + m];
  float p[3], q1[3], q2[3], d1[5], d2[5];
  #pragma unroll
  for (int j=0;j<3;++j){
    p[j]  = f1[bf1 + 256  + 3*m + j];
    q1[j] = f2[bf2 +        3*m + j];
    q2[j] = f2[bf2 + 768  + 3*m + j];
  }
  #pragma unroll
  for (int j=0;j<5;++j){
    d1[j] = f1[bf1 + 1024 + 5*m + j];
    d2[j] = f2[bf2 + 1536 + 5*m + j];
  }

  // pair 0 (l1=0): plain broadcast product, no CG
  #pragma unroll
  for (int j=0;j<3;++j) out[bo + 3*m + j] = s*q1[j];

  // stage K-vector: [ p (x) q2 | d1 (x) d2 | 0 0 ]
  float u[NROW];
  #pragma unroll
  for (int j=0;j<3;++j)
    #pragma unroll
    for (int k=0;k<3;++k) u[3*j+k] = p[j]*q2[k];
  #pragma unroll
  for (int j=0;j<5;++j)
    #pragma unroll
    for (int k=0;k<5;++k) u[9+5*j+k] = d1[j]*d2[k];
  u[34] = 0.f; u[35] = 0.f;

  #pragma unroll
  for (int j=0;j<NROW;j+=4){
    v4f t = {u[j],u[j+1],u[j+2],u[j+3]};
    *(v4f*)&lds[m*NROW + j] = t;          // 16B aligned: 144*m + 16*j/4
  }
  __syncthreads();

  // A-fragment (16x4 f32): lanes 0-15 row M=l, K+0/K+1; lanes 16-31 row M=l-16, K+2/K+3
  v8f acc[2][3] = {};
  const int khalf = (lane >> 4) << 1;
  const int nlow  = lane & 15;
  const float* a0p = &lds[(wbase + nlow)*NROW + khalf];
  const float* a1p = a0p + 16*NROW;
  #pragma unroll
  for (int kk=0; kk<9; ++kk){
    v2f a0 = *(const v2f*)(a0p + 4*kk);
    v2f a1 = *(const v2f*)(a1p + 4*kk);
    #pragma unroll
    for (int ni=0; ni<3; ++ni){
      if (ni==0 && kk>2) continue;  // pair-1 cols: K 0..11 only (block diag)
      if (ni!=0 && kk<2) continue;  // pair-2 cols: K 8..35 only
      // B-fragment (4x16 f32), column-major W makes it the dual of A
      v2f bf = *(const v2f*)&Wcm[(ni*16 + nlow)*NROW + 4*kk + khalf];
      acc[0][ni] = __builtin_amdgcn_wmma_f32_16x16x4_f32(
          false, a0, false, bf, (short)0, acc[0][ni], false, false);
      acc[1][ni] = __builtin_amdgcn_wmma_f32_16x16x4_f32(
          false, a1, false, bf, (short)0, acc[1][ni], false, false);
    }
  }

  // D layout: VGPR v, lanes 0-15 -> row v, lanes 16-31 -> row v+8; col = lane&15
  const int mhi = (lane >> 4) << 3;
  #pragma unroll
  for (int ni=0; ni<3; ++ni){
    int col = ni*16 + nlow;
    if (col < 34){
      int off = bo + cOff[col];
      int str = cStr[col];
      #pragma unroll
      for (int mi=0; mi<2; ++mi){
        int mr = wbase + mi*16 + mhi;
        #pragma unroll
        for (int v=0; v<8; ++v)
          out[off + (mr+v)*str] = acc[mi][ni][v];
      }
    }
  }
}

extern "C" void kernel_launch(void* const* d_in, const int* in_sizes, int n_in,
                              void* d_out, int out_size, void* d_ws, size_t ws_size,
                              hipStream_t stream){
  const float* f1 = (const float*)d_in[0];
  const float* f2 = (const float*)d_in[1];
  float* out = (float*)d_out;
  float* Wcm = (float*)d_ws;                 // 36*48 floats = 6912 B of scratch
  int B = in_sizes[0] / 2304;
  build_w<<<(NROW*NCOL + 255)/256, 256, 0, stream>>>(Wcm);
  tp_wmma_kernel<<<B, 256, 0, stream>>>(f1, f2, Wcm, out);
}